// LoonUNet_58162447123329
// MI455X (gfx1250) — compile-verified
//
#include <hip/hip_runtime.h>
#include <hip/hip_bf16.h>

typedef float v2f __attribute__((ext_vector_type(2)));
typedef float v8f __attribute__((ext_vector_type(8)));

#define GN_EPS 1e-5f
#define KNN_K 16

// ---------------------------------------------------------------------------
// Simple elementwise helpers
// ---------------------------------------------------------------------------
__global__ void copy_f(const float* __restrict__ src, float* __restrict__ dst, int n) {
    int i = blockIdx.x * blockDim.x + threadIdx.x;
    if (i < n) dst[i] = src[i];
}

// F0[b,o,n] = sum_c stem_w[o,c]*xyz[b,c,n] + stem_b[o]   (8 out channels)
__global__ void stem_kernel(const float* __restrict__ xyz, const float* __restrict__ w,
                            const float* __restrict__ bias, float* __restrict__ F0,
                            int B, int N) {
    int i = blockIdx.x * blockDim.x + threadIdx.x;
    int total = B * 8 * N;
    if (i >= total) return;
    int n = i % N;
    int o = (i / N) % 8;
    int b = i / (8 * N);
    const float* x = xyz + (size_t)b * 3 * N;
    F0[i] = w[o * 3 + 0] * x[n] + w[o * 3 + 1] * x[N + n] + w[o * 3 + 2] * x[2 * N + n] + bias[o];
}

// dst[b,c,m] = src[b,c,idx[b,m]]
__global__ void gather_kernel(const float* __restrict__ src, const int* __restrict__ idx,
                              float* __restrict__ dst, int B, int C, int Ns, int M) {
    int i = blockIdx.x * blockDim.x + threadIdx.x;
    int total = B * C * M;
    if (i >= total) return;
    int m = i % M;
    int c = (i / M) % C;
    int b = i / (M * C);
    dst[i] = src[((size_t)b * C + c) * Ns + idx[(size_t)b * M + m]];
}

// ---------------------------------------------------------------------------
// KNN via V_WMMA_F32_16X16X4_F32.
// One wave handles 32 queries (two 16-row WMMA tiles). Per 16-candidate tile
// two WMMAs compute  score[m,n] = -2 q_m . p_n + |p_n|^2  (the per-query
// |q|^2 constant cannot change each row's top-k ordering, so it is dropped).
// A (16x4): lane l<16 -> M=l : V0=-2qx, V1=-2qy ; lane 16+l : V0=-2qz, V1=1
// B (4x16): lane l<16 -> N=l : V0=px,  V1=py  ; lane 16+l : V0=pz,  V1=|p|^2
// C (16x16): VGPR j, lanes0-15 -> (M=j,N=lane) ; lanes16-31 -> (M=j+8,N=lane-16)
// Scores go to LDS sd[32][17]; then EVERY lane owns one query row and keeps a
// register-resident sorted top-16 updated with a branch-free select chain.
// ---------------------------------------------------------------------------
__global__ __launch_bounds__(32) void knn_wmma_kernel(const float* __restrict__ Q,
                                                      const float* __restrict__ P,
                                                      int* __restrict__ idx_out,
                                                      int Nq, int Np) {
    int tilesPerB = Nq >> 5;            // 32 queries per wave
    int b = blockIdx.x / tilesPerB;
    int qt = blockIdx.x % tilesPerB;
    int lane = threadIdx.x;
    int half = lane >> 4;
    int l = lane & 15;
    int m0 = qt << 5;

    const float* Qb = Q + (size_t)b * 3 * Nq;
    const float* Pb = P + (size_t)b * 3 * Np;

    // A operands for the two query tiles (rows m0+l and m0+16+l)
    int ma = m0 + l, mb = m0 + 16 + l;
    float qax = Qb[ma], qay = Qb[Nq + ma], qaz = Qb[2 * Nq + ma];
    float qbx = Qb[mb], qby = Qb[Nq + mb], qbz = Qb[2 * Nq + mb];
    v2f A0, A1;
    if (half == 0) { A0.x = -2.f * qax; A0.y = -2.f * qay;
                     A1.x = -2.f * qbx; A1.y = -2.f * qby; }
    else           { A0.x = -2.f * qaz; A0.y = 1.f;
                     A1.x = -2.f * qbz; A1.y = 1.f; }

    __shared__ float sd[32][17];

    float bd[16];
    int bi[16];
#pragma unroll
    for (int i = 0; i < 16; ++i) { bd[i] = 3.0e38f; bi[i] = 0; }

    for (int nt = 0; nt < Np; nt += 16) {
        int nn = nt + l;
        float px = Pb[nn], py = Pb[Np + nn], pz = Pb[2 * Np + nn];
        v2f Bv;
        if (half == 0) { Bv.x = px; Bv.y = py; }
        else           { Bv.x = pz; Bv.y = px * px + py * py + pz * pz; }

        v8f C0 = {};
        v8f C1 = {};
        C0 = __builtin_amdgcn_wmma_f32_16x16x4_f32(false, A0, false, Bv,
                                                   (short)0, C0, false, false);
        C1 = __builtin_amdgcn_wmma_f32_16x16x4_f32(false, A1, false, Bv,
                                                   (short)0, C1, false, false);
#pragma unroll
        for (int j = 0; j < 8; ++j) {
            sd[j + 8 * half][l]      = C0[j];
            sd[16 + j + 8 * half][l] = C1[j];
        }
        __syncthreads();

        // every lane owns query row m0+lane
#pragma unroll
        for (int c = 0; c < 16; ++c) {
            float d = sd[lane][c];
            if (d < bd[15]) {                 // pure fast-path guard
                float cd = d; int ci = nt + c;
#pragma unroll
                for (int i = 0; i < 16; ++i) {
                    bool sw = bd[i] > cd;     // select-based carry insert
                    float tb = bd[i]; int ti = bi[i];
                    bd[i] = sw ? cd : tb;
                    bi[i] = sw ? ci : ti;
                    cd = sw ? tb : cd;
                    ci = sw ? ti : ci;
                }
            }
        }
        __syncthreads();
    }

    size_t base = ((size_t)b * Nq + m0 + lane) * KNN_K;
#pragma unroll
    for (int i = 0; i < 16; ++i) idx_out[base + i] = bi[i];
}

// ---------------------------------------------------------------------------
// EdgeConv: gather + concat(nbr-Fi, Fi) + 1x1 conv (WMMA f32 16x16x4) +
// GroupNorm(8 groups) + ReLU + max over K neighbors.
// One wave per query point. pair[2Cin][16] staged in LDS; per 16-output tile
// 2Cin/4 accumulating WMMAs (A = W tile, B = pair tile, N dim = 16 neighbors).
// pass 0: write per-block group sum/sumsq partials (deterministic reduce later)
// pass 1: recompute conv, normalize with (mu, rinv), ReLU, shfl-max over K.
// ---------------------------------------------------------------------------
__global__ __launch_bounds__(32) void edgeconv_kernel(const float* __restrict__ Fq,
                                                      const float* __restrict__ Fk,
                                                      const int* __restrict__ nidx,
                                                      const float* __restrict__ W,
                                                      const float* __restrict__ gamma,
                                                      const float* __restrict__ beta,
                                                      const float* __restrict__ stats,
                                                      float* __restrict__ part,
                                                      float* __restrict__ out,
                                                      int Nq, int Np, int Cin, int Cout,
                                                      int pass) {
    int blk = blockIdx.x;
    int b = blk / Nq;
    int n = blk % Nq;
    int lane = threadIdx.x;
    int half = lane >> 4;
    int l = lane & 15;

    __shared__ float spair[128 * 16];   // up to 2Cin=128 channels x 16 neighbors
    __shared__ int sidx[16];
    __shared__ float s_g[8], s_q[8], s_mu[8], s_ri[8];

    if (lane < 16) sidx[lane] = nidx[((size_t)b * Nq + n) * KNN_K + lane];
    if (pass == 0) {
        if (lane < 8) { s_g[lane] = 0.f; s_q[lane] = 0.f; }
    } else {
        if (lane < 8) {
            s_mu[lane] = stats[32 + b * 8 + lane];
            s_ri[lane] = stats[48 + b * 8 + lane];
        }
    }
    __syncthreads();

    // stage pair = [nbr - Fi ; Fi], layout spair[c][k]
    for (int t = lane; t < Cin * 16; t += 32) {
        int c = t >> 4, k = t & 15;
        float fi = Fq[((size_t)b * Cin + c) * Nq + n];
        float nb = Fk[((size_t)b * Cin + c) * Np + sidx[k]];
        spair[c * 16 + k] = nb - fi;
        spair[(Cin + c) * 16 + k] = fi;
    }
    __syncthreads();

    const int C2 = 2 * Cin;
    const int cpg = Cout >> 3;   // channels per group (G = 8)

    for (int m0 = 0; m0 < Cout; m0 += 16) {
        v8f acc = {};
        int row = m0 + l;
        for (int c0 = 0; c0 < C2; c0 += 4) {
            v2f Av, Bv;
            Av.x = W[(size_t)row * C2 + c0 + 2 * half];
            Av.y = W[(size_t)row * C2 + c0 + 2 * half + 1];
            Bv.x = spair[(c0 + 2 * half) * 16 + l];
            Bv.y = spair[(c0 + 2 * half + 1) * 16 + l];
            acc = __builtin_amdgcn_wmma_f32_16x16x4_f32(false, Av, false, Bv,
                                                        (short)0, acc, false, false);
        }
        int obase = m0 + 8 * half;   // this lane's channels are obase..obase+7
        if (pass == 0) {
            float s0 = 0.f, q0 = 0.f, s1 = 0.f, q1 = 0.f;
#pragma unroll
            for (int j = 0; j < 4; ++j) { float v = acc[j]; s0 += v; q0 += v * v; }
#pragma unroll
            for (int j = 4; j < 8; ++j) { float v = acc[j]; s1 += v; q1 += v * v; }
            atomicAdd(&s_g[obase / cpg], s0);
            atomicAdd(&s_q[obase / cpg], q0);
            atomicAdd(&s_g[(obase + 4) / cpg], s1);
            atomicAdd(&s_q[(obase + 4) / cpg], q1);
        } else {
#pragma unroll
            for (int j = 0; j < 8; ++j) {
                int o = obase + j;
                int g = o / cpg;
                float y = (acc[j] - s_mu[g]) * s_ri[g] * gamma[o] + beta[o];
                y = fmaxf(y, 0.f);
                // max over K=16 neighbors (lanes within each 16-lane half)
#pragma unroll
                for (int sft = 1; sft < 16; sft <<= 1)
                    y = fmaxf(y, __shfl_xor(y, sft, 32));
                if (l == 0) out[((size_t)b * Cout + o) * Nq + n] = y;
            }
        }
    }

    if (pass == 0) {
        __syncthreads();
        if (lane < 8) {
            part[(size_t)blk * 16 + lane] = s_g[lane];
            part[(size_t)blk * 16 + 8 + lane] = s_q[lane];
        }
    }
}

// Deterministic fixed-order reduction of per-block GroupNorm partials.
// grid = B*16 blocks: blockIdx = b*16 + j, j<8 -> sum, j>=8 -> sumsq.
__global__ void gn_reduce(const float* __restrict__ part, float* __restrict__ stats, int Nq) {
    int b = blockIdx.x >> 4;
    int j = blockIdx.x & 15;
    int t = threadIdx.x;
    __shared__ float sv[256];
    float s = 0.f;
    for (int n = t; n < Nq; n += 256) s += part[((size_t)(b * Nq + n)) * 16 + j];
    sv[t] = s;
    __syncthreads();
    for (int off = 128; off > 0; off >>= 1) {
        if (t < off) sv[t] += sv[t + off];
        __syncthreads();
    }
    if (t == 0) {
        if (j < 8) stats[b * 8 + j] = sv[0];
        else       stats[16 + b * 8 + (j - 8)] = sv[0];
    }
}

// mu = sum/cnt ; rinv = rsqrt(sumsq/cnt - mu^2 + eps)
__global__ void gn_finalize(float* __restrict__ stats, int ng, float cnt) {
    int i = threadIdx.x;
    if (i < ng) {
        float s = stats[i];
        float q = stats[16 + i];
        float mu = s / cnt;
        float var = q / cnt - mu * mu;
        stats[32 + i] = mu;
        stats[48 + i] = rsqrtf(var + GN_EPS);
    }
}

// ---------------------------------------------------------------------------
// Farthest point sampling, start at index 0, argmax ties -> lowest index
// (matches jnp.argmax). One 256-thread block per batch.
// ---------------------------------------------------------------------------
__global__ void fps_kernel(const float* __restrict__ P, int N, int M,
                           int* __restrict__ idx_out, float* __restrict__ dists) {
    int b = blockIdx.x;
    const float* Pb = P + (size_t)b * 3 * N;
    float* db = dists + (size_t)b * N;
    int t = threadIdx.x;
    const int nt = 256;
    __shared__ float s_v[256];
    __shared__ int s_i[256];
    __shared__ int s_cur;

    float x0 = Pb[0], y0 = Pb[N], z0 = Pb[2 * N];
    for (int i = t; i < N; i += nt) {
        float dx = Pb[i] - x0, dy = Pb[N + i] - y0, dz = Pb[2 * N + i] - z0;
        db[i] = dx * dx + dy * dy + dz * dz;
    }
    if (t == 0) idx_out[(size_t)b * M] = 0;
    __syncthreads();

    for (int s = 1; s < M; ++s) {
        float best = -1.f;
        int bix = 0x7fffffff;
        for (int i = t; i < N; i += nt) {
            float d = db[i];
            if (d > best) { best = d; bix = i; }
        }
        s_v[t] = best; s_i[t] = bix;
        __syncthreads();
        for (int off = 128; off > 0; off >>= 1) {
            if (t < off) {
                float ov = s_v[t + off];
                int oi = s_i[t + off];
                if (ov > s_v[t] || (ov == s_v[t] && oi < s_i[t])) { s_v[t] = ov; s_i[t] = oi; }
            }
            __syncthreads();
        }
        if (t == 0) { s_cur = s_i[0]; idx_out[(size_t)b * M + s] = s_i[0]; }
        __syncthreads();
        int nxt = s_cur;
        float px = Pb[nxt], py = Pb[N + nxt], pz = Pb[2 * N + nxt];
        for (int i = t; i < N; i += nt) {
            float dx = Pb[i] - px, dy = Pb[N + i] - py, dz = Pb[2 * N + i] - pz;
            float nd = dx * dx + dy * dy + dz * dz;
            if (nd < db[i]) db[i] = nd;
        }
        __syncthreads();
    }
}

// ---------------------------------------------------------------------------
// Orchestration
// ---------------------------------------------------------------------------
extern "C" void kernel_launch(void* const* d_in, const int* in_sizes, int n_in,
                              void* d_out, int out_size, void* d_ws, size_t ws_size,
                              hipStream_t stream) {
    const float* xyz    = (const float*)d_in[0];
    const float* stem_w = (const float*)d_in[1];
    const float* stem_b = (const float*)d_in[2];
    const float* w1 = (const float*)d_in[3];
    const float* g1 = (const float*)d_in[4];
    const float* b1 = (const float*)d_in[5];
    const float* w2 = (const float*)d_in[6];
    const float* g2 = (const float*)d_in[7];
    const float* b2 = (const float*)d_in[8];
    const float* w3 = (const float*)d_in[9];
    const float* g3 = (const float*)d_in[10];
    const float* b3 = (const float*)d_in[11];
    const float* w4 = (const float*)d_in[12];
    const float* g4 = (const float*)d_in[13];
    const float* b4 = (const float*)d_in[14];

    const int B = 2;
    const int N0 = in_sizes[0] / (3 * B);   // 8192
    const int N1 = N0 / 4;                  // 2048
    const int N2 = N1 / 4;                  // 512

    // output segments (return order: P0, F0a, P1, F1a, P2, F2a)
    float* out = (float*)d_out;
    float* P0o = out;
    float* F0a = P0o + (size_t)B * 3 * N0;
    float* P1o = F0a + (size_t)B * 32 * N0;
    float* F1a = P1o + (size_t)B * 3 * N1;
    float* P2o = F1a + (size_t)B * 64 * N1;
    float* F2a = P2o + (size_t)B * 3 * N2;

    // workspace carve-up
    float* wsf = (float*)d_ws;
    float* F0    = wsf; wsf += (size_t)B * 8 * N0;
    float* F1sk  = wsf; wsf += (size_t)B * 32 * N1;
    float* F2sk  = wsf; wsf += (size_t)B * 64 * N2;
    float* F2mid = wsf; wsf += (size_t)B * 64 * N2;
    float* fpsd  = wsf; wsf += (size_t)B * N0;
    float* stats = wsf; wsf += 64;                       // sum[16] sq[16] mu[16] rinv[16]
    float* part  = wsf; wsf += (size_t)B * N0 * 16;      // per-block GN partials
    int* knn_idx = (int*)wsf; wsf += (size_t)B * N0 * KNN_K;
    int* idx1    = (int*)wsf; wsf += (size_t)B * N1;
    int* idx2    = (int*)wsf; wsf += (size_t)B * N2;

    auto cdiv = [](int a, int b) { return (a + b - 1) / b; };

    auto run_ec = [&](const float* Fq, const float* Fk, const int* nidx,
                      const float* Wm, const float* ga, const float* be,
                      float* dst, int Cin, int Cout, int Nq, int Np) {
        edgeconv_kernel<<<B * Nq, 32, 0, stream>>>(Fq, Fk, nidx, Wm, ga, be, stats,
                                                   part, dst, Nq, Np, Cin, Cout, 0);
        gn_reduce<<<B * 16, 256, 0, stream>>>(part, stats, Nq);
        float cnt = (float)(((size_t)(Cout / 8)) * (size_t)Nq * KNN_K);
        gn_finalize<<<1, 32, 0, stream>>>(stats, B * 8, cnt);
        edgeconv_kernel<<<B * Nq, 32, 0, stream>>>(Fq, Fk, nidx, Wm, ga, be, stats,
                                                   part, dst, Nq, Np, Cin, Cout, 1);
    };

    // P0 = xyz
    { int n = B * 3 * N0; copy_f<<<cdiv(n, 256), 256, 0, stream>>>(xyz, P0o, n); }
    // stem: F0 = stem_w @ xyz + stem_b
    { int n = B * 8 * N0; stem_kernel<<<cdiv(n, 256), 256, 0, stream>>>(xyz, stem_w, stem_b, F0, B, N0); }

    // stage 0: EdgeConv(F0 | P0,P0) -> F0a [B,32,N0]
    knn_wmma_kernel<<<B * (N0 / 32), 32, 0, stream>>>(P0o, P0o, knn_idx, N0, N0);
    run_ec(F0, F0, knn_idx, w1, g1, b1, F0a, 8, 32, N0, N0);

    // FPS -> N1, gather P1 and skip features
    fps_kernel<<<B, 256, 0, stream>>>(P0o, N0, N1, idx1, fpsd);
    gather_kernel<<<cdiv(B * 3 * N1, 256), 256, 0, stream>>>(P0o, idx1, P1o, B, 3, N0, N1);
    gather_kernel<<<cdiv(B * 32 * N1, 256), 256, 0, stream>>>(F0a, idx1, F1sk, B, 32, N0, N1);

    // stage 1: EdgeConv(F1sk,F0a | P1,P0) -> F1a [B,64,N1]
    knn_wmma_kernel<<<B * (N1 / 32), 32, 0, stream>>>(P1o, P0o, knn_idx, N1, N0);
    run_ec(F1sk, F0a, knn_idx, w2, g2, b2, F1a, 32, 64, N1, N0);

    // FPS -> N2, gather P2 and skip features
    fps_kernel<<<B, 256, 0, stream>>>(P1o, N1, N2, idx2, fpsd);
    gather_kernel<<<cdiv(B * 3 * N2, 256), 256, 0, stream>>>(P1o, idx2, P2o, B, 3, N1, N2);
    gather_kernel<<<cdiv(B * 64 * N2, 256), 256, 0, stream>>>(F1a, idx2, F2sk, B, 64, N1, N2);

    // stage 2: two EdgeConvs share the same (P2,P1) KNN graph
    knn_wmma_kernel<<<B * (N2 / 32), 32, 0, stream>>>(P2o, P1o, knn_idx, N2, N1);
    run_ec(F2sk,  F1a, knn_idx, w3, g3, b3, F2mid, 64, 64,  N2, N1);
    run_ec(F2mid, F1a, knn_idx, w4, g4, b4, F2a,   64, 128, N2, N1);
}